// LSTMTagger_85787676771235
// MI455X (gfx1250) — compile-verified
//
#include <hip/hip_runtime.h>
#include <math.h>

// ---- problem dims (match reference) ----
#define S_WORDS 8192
#define WL      16
#define EW      256
#define EC      32
#define HC      64
#define HW      512
#define T_TAGS  50
#define GC      (4*HC)   // 256 char gate width
#define GW      (4*HW)   // 2048 word gate width
#define KX      (EW+HC)  // 320 word-LSTM input dim

typedef float v2f __attribute__((ext_vector_type(2)));
typedef float v8f __attribute__((ext_vector_type(8)));

__device__ __forceinline__ float sigf(float x) { return 1.0f / (1.0f + __expf(-x)); }

// D(16x16,f32) += A(16x4,f32) * B(4x16,f32)   -- V_WMMA_F32_16X16X4_F32
__device__ __forceinline__ v8f wmma4(v2f a, v2f b, v8f c) {
  return __builtin_amdgcn_wmma_f32_16x16x4_f32(false, a, false, b, (short)0, c, false, false);
}

// =====================================================================
// Kernel 1: char-level LSTM, batched over all S words via f32 WMMA.
// Weights live in LDS (padded rows to break bank conflicts); h/c state
// stays in D-layout VGPRs across the 16 timesteps; h is transposed
// through LDS for the A operand. Dynamic LDS (~117KB) so the >64KB
// allocation bypasses static limits (WGP allows 320KB).
// =====================================================================
#define CW    4              // waves per block
#define WIH_P 34             // padded row stride (floats) for Wih_c [256][32]
#define WHH_P 66             // padded row stride for Whh_c [256][64]
#define HST_P 66             // padded row stride for h staging [16][64]
#define CH_LDS_FLOATS (256*WIH_P + 256*WHH_P + CW*16*HST_P)
#define CH_LDS_BYTES  (CH_LDS_FLOATS*4 + (CW*16)*4)

__global__ __launch_bounds__(CW*32)
void char_lstm_kernel(const int* __restrict__ chars,
                      const int* __restrict__ char_lengths,
                      const float* __restrict__ char_emb,
                      const float* __restrict__ Wih_c,
                      const float* __restrict__ Whh_c,
                      const float* __restrict__ bih_c,
                      const float* __restrict__ bhh_c,
                      float* __restrict__ h_char) {
  extern __shared__ char smem_raw[];
  float* lds_wih = (float*)smem_raw;                 // [256][WIH_P]
  float* lds_whh = lds_wih + 256 * WIH_P;            // [256][WHH_P]
  float* hstage  = lds_whh + 256 * WHH_P;            // [CW][16*HST_P]
  int*   lens    = (int*)(hstage + CW * 16 * HST_P); // [CW*16]

  const int tid  = threadIdx.x;
  const int wave = tid >> 5, lane = tid & 31;
  const int row  = lane & 15;          // A-row / B,C,D column index for this lane
  const int hi   = lane >> 4;          // 0: lanes 0-15, 1: lanes 16-31
  const int wg0  = blockIdx.x * (CW * 16);
  const int wbase = wg0 + wave * 16;

  // cooperative weight staging into padded LDS
  for (int i = tid; i < 256 * EC; i += CW * 32)
    lds_wih[(i >> 5) * WIH_P + (i & 31)] = Wih_c[i];
  for (int i = tid; i < 256 * HC; i += CW * 32)
    lds_whh[(i >> 6) * WHH_P + (i & 63)] = Whh_c[i];
  if (tid < CW * 16) lens[tid] = char_lengths[wg0 + tid];
  __syncthreads();

  // bias per (gate, jt) for this lane's column j = jt*16 + row
  float bsum[4][4];
  for (int g = 0; g < 4; ++g)
    for (int jt = 0; jt < 4; ++jt) {
      int col = g * 64 + jt * 16 + row;
      bsum[g][jt] = bih_c[col] + bhh_c[col];
    }

  // h/c state in C/D layout: element (m = r + 8*hi, j = jt*16 + row)
  v8f h[4], c[4];
  for (int jt = 0; jt < 4; ++jt)
    for (int e = 0; e < 8; ++e) { h[jt][e] = 0.f; c[jt][e] = 0.f; }

  float* hst = hstage + wave * 16 * HST_P;

  for (int t = 0; t < WL; ++t) {
    // stage h (padded row-major [m][j]) so it can be re-read in A layout
#pragma unroll
    for (int jt = 0; jt < 4; ++jt)
#pragma unroll
      for (int r = 0; r < 8; ++r)
        hst[(r + 8 * hi) * HST_P + jt * 16 + row] = h[jt][r];
    __syncthreads();

    const int cid = chars[(size_t)(wbase + row) * WL + t];
    const float* ce   = char_emb + (size_t)cid * EC;
    const float* hrow = hst + row * HST_P;

#pragma unroll
    for (int jt = 0; jt < 4; ++jt) {
      // four gate tiles (i,f,g,o) for column group jt share the A operand
      v8f acc[4];
      for (int g = 0; g < 4; ++g)
        for (int e = 0; e < 8; ++e) acc[g][e] = 0.f;
      const int n0 = jt * 16 + row;  // column within a 64-wide gate block

      // x @ Wih_c^T   (K = 32)
#pragma unroll
      for (int kk = 0; kk < EC; kk += 4) {
        v2f a = *(const v2f*)(ce + kk + 2 * hi);
#pragma unroll
        for (int g = 0; g < 4; ++g) {
          v2f b = *(const v2f*)(lds_wih + (g * 64 + n0) * WIH_P + kk + 2 * hi);
          acc[g] = wmma4(a, b, acc[g]);
        }
      }
      // h @ Whh_c^T   (K = 64)
#pragma unroll
      for (int kk = 0; kk < HC; kk += 4) {
        v2f a = *(const v2f*)(hrow + kk + 2 * hi);
#pragma unroll
        for (int g = 0; g < 4; ++g) {
          v2f b = *(const v2f*)(lds_whh + (g * 64 + n0) * WHH_P + kk + 2 * hi);
          acc[g] = wmma4(a, b, acc[g]);
        }
      }
      // gates + masked state update (mask: t < char_lengths[word m])
#pragma unroll
      for (int r = 0; r < 8; ++r) {
        int m = r + 8 * hi;
        if (t < lens[wave * 16 + m]) {
          float iv = sigf(acc[0][r] + bsum[0][jt]);
          float fv = sigf(acc[1][r] + bsum[1][jt]);
          float gv = tanhf(acc[2][r] + bsum[2][jt]);
          float ov = sigf(acc[3][r] + bsum[3][jt]);
          float cv = fv * c[jt][r] + iv * gv;
          c[jt][r] = cv;
          h[jt][r] = ov * tanhf(cv);
        }
      }
    }
    __syncthreads();  // protect hstage before next iteration's overwrite
  }

  // write final h_char [S, HC]
  for (int jt = 0; jt < 4; ++jt)
    for (int r = 0; r < 8; ++r) {
      int m = r + 8 * hi;
      h_char[(size_t)(wbase + m) * HC + jt * 16 + row] = h[jt][r];
    }
}

// =====================================================================
// Kernel 2: hoisted input-to-hidden GEMM for the word LSTM:
// precomp[S, 2048] = [word_emb[sentence] || h_char] @ Wih_w^T + bih + bhh
// M=8192, N=2048, K=320, f32 WMMA. Two independent 16x16 accumulator
// chains per wave (shared A) for ILP across the wmma latency.
// =====================================================================
__global__ __launch_bounds__(128)
void wih_precompute_kernel(const int* __restrict__ sentence,
                           const float* __restrict__ word_emb,
                           const float* __restrict__ h_char,
                           const float* __restrict__ Wih_w,
                           const float* __restrict__ bih_w,
                           const float* __restrict__ bhh_w,
                           float* __restrict__ precomp) {
  const int tid  = threadIdx.x;
  const int wave = tid >> 5, lane = tid & 31;
  const int row  = lane & 15, hi = lane >> 4;
  const int m0 = blockIdx.x * 16;
  const int n0 = blockIdx.y * 128 + wave * 32;
  const int na = n0 + row, nb = n0 + 16 + row;

  const float* we  = word_emb + (size_t)sentence[m0 + row] * EW;  // x-row (k<256)
  const float* hc  = h_char + (size_t)(m0 + row) * HC;            // x-row (k>=256)
  const float* wra = Wih_w + (size_t)na * KX;
  const float* wrb = Wih_w + (size_t)nb * KX;

  v8f acc0, acc1;
  for (int e = 0; e < 8; ++e) { acc0[e] = 0.f; acc1[e] = 0.f; }

#pragma unroll 8
  for (int kk = 0; kk < EW; kk += 4) {
    v2f a  = *(const v2f*)(we + kk + 2 * hi);
    v2f b0 = *(const v2f*)(wra + kk + 2 * hi);
    v2f b1 = *(const v2f*)(wrb + kk + 2 * hi);
    acc0 = wmma4(a, b0, acc0);
    acc1 = wmma4(a, b1, acc1);
  }
#pragma unroll
  for (int kk = 0; kk < HC; kk += 4) {
    v2f a  = *(const v2f*)(hc + kk + 2 * hi);
    v2f b0 = *(const v2f*)(wra + EW + kk + 2 * hi);
    v2f b1 = *(const v2f*)(wrb + EW + kk + 2 * hi);
    acc0 = wmma4(a, b0, acc0);
    acc1 = wmma4(a, b1, acc1);
  }

  const float bba = bih_w[na] + bhh_w[na];
  const float bbb = bih_w[nb] + bhh_w[nb];
#pragma unroll
  for (int r = 0; r < 8; ++r) {
    int m = m0 + r + 8 * hi;
    precomp[(size_t)m * GW + na] = acc0[r] + bba;
    precomp[(size_t)m * GW + nb] = acc1[r] + bbb;
  }
}

// =====================================================================
// Kernel 3: persistent word-LSTM recurrence. Whh (2048x512 f32 = 4MB)
// lives entirely in VGPRs: 32 WGs x 512 thr x 64 weights. Per step:
// agent-scope h broadcast through L2, 64 FMAs/thread, LDS reduce,
// global counter barrier (double-buffered h to avoid WAR races).
// global_prefetch of next step's precomp row overlaps the L2 read.
// =====================================================================
#define NWG    32
#define HSLICE 16   // h elements owned per WG
#define ROWS   64   // gate rows per WG (4 gates x 16)
#define SEGLEN 64   // k-chunk per thread (8 threads per row)

__global__ __launch_bounds__(512)
void word_lstm_kernel(const float* __restrict__ Whh_w,
                      const float* __restrict__ precomp,
                      float* __restrict__ lstm_out,
                      float* __restrict__ h_buf,          // [2][HW]
                      unsigned int* __restrict__ counter) {
  __shared__ float hprev[HW];
  __shared__ float partial[512];
  __shared__ float gact[ROWS];
  __shared__ float cst[HSLICE];

  const int wg = blockIdx.x, tid = threadIdx.x;
  const int rl = tid >> 3, seg = tid & 7;       // 64 rows x 8 segments
  const int gate = rl >> 4, j = rl & 15;
  const int hbase = wg * HSLICE;
  const int grow = gate * HW + hbase + j;       // global gate row

  float w[SEGLEN];
#pragma unroll
  for (int k = 0; k < SEGLEN; ++k)
    w[k] = Whh_w[(size_t)grow * HW + seg * SEGLEN + k];

  if (tid < HSLICE) cst[tid] = 0.f;
  __syncthreads();

  for (int t = 0; t < S_WORDS; ++t) {
    const int rb = t & 1;
    if (tid < 128) {
#pragma unroll
      for (int q = 0; q < 4; ++q)
        hprev[tid * 4 + q] = __hip_atomic_load(&h_buf[rb * HW + tid * 4 + q],
                                               __ATOMIC_RELAXED,
                                               __HIP_MEMORY_SCOPE_AGENT);
    }
    // prefetch next step's precomp slice (gfx1250 global_prefetch_b8)
    if (tid < ROWS && t + 1 < S_WORDS)
      __builtin_prefetch(&precomp[(size_t)(t + 1) * GW + (tid >> 4) * HW +
                                  hbase + (tid & 15)], 0, 0);
    __syncthreads();

    float s = 0.f;
    const float* hp = &hprev[seg * SEGLEN];
#pragma unroll
    for (int k = 0; k < SEGLEN; ++k) s = fmaf(w[k], hp[k], s);
    partial[tid] = s;
    __syncthreads();

    if (tid < ROWS) {
      float g = partial[tid * 8 + 0] + partial[tid * 8 + 1] +
                partial[tid * 8 + 2] + partial[tid * 8 + 3] +
                partial[tid * 8 + 4] + partial[tid * 8 + 5] +
                partial[tid * 8 + 6] + partial[tid * 8 + 7];
      gact[tid] = g + precomp[(size_t)t * GW + (tid >> 4) * HW + hbase + (tid & 15)];
    }
    __syncthreads();

    if (tid < HSLICE) {
      float iv = sigf(gact[tid]);
      float fv = sigf(gact[16 + tid]);
      float gv = tanhf(gact[32 + tid]);
      float ov = sigf(gact[48 + tid]);
      float cv = fv * cst[tid] + iv * gv;
      cst[tid] = cv;
      float hv = ov * tanhf(cv);
      lstm_out[(size_t)t * HW + hbase + tid] = hv;
      __hip_atomic_store(&h_buf[(rb ^ 1) * HW + hbase + tid], hv,
                         __ATOMIC_RELAXED, __HIP_MEMORY_SCOPE_AGENT);
    }
    __syncthreads();

    if (tid == 0) {
      __threadfence();
      __hip_atomic_fetch_add(counter, 1u, __ATOMIC_RELEASE,
                             __HIP_MEMORY_SCOPE_AGENT);
      const unsigned int target = (unsigned int)(t + 1) * NWG;
      while (__hip_atomic_load(counter, __ATOMIC_ACQUIRE,
                               __HIP_MEMORY_SCOPE_AGENT) < target)
        __builtin_amdgcn_s_sleep(2);
    }
    __syncthreads();
  }
}

// =====================================================================
// Kernel 4: tag projection + log_softmax (one block per word).
// =====================================================================
__global__ __launch_bounds__(64)
void tag_kernel(const float* __restrict__ lstm_out,
                const float* __restrict__ W_tag,
                const float* __restrict__ b_tag,
                float* __restrict__ out) {
  __shared__ float hrow[HW];
  __shared__ float sc[T_TAGS];
  __shared__ float red[2];
  const int r = blockIdx.x, tid = threadIdx.x;

  for (int k = tid; k < HW; k += 64) hrow[k] = lstm_out[(size_t)r * HW + k];
  __syncthreads();

  if (tid < T_TAGS) {
    float s = b_tag[tid];
    const float* wt = W_tag + (size_t)tid * HW;
#pragma unroll 8
    for (int k = 0; k < HW; ++k) s = fmaf(wt[k], hrow[k], s);
    sc[tid] = s;
  }
  __syncthreads();
  if (tid == 0) {
    float m = sc[0];
    for (int i = 1; i < T_TAGS; ++i) m = fmaxf(m, sc[i]);
    float se = 0.f;
    for (int i = 0; i < T_TAGS; ++i) se += __expf(sc[i] - m);
    red[0] = m; red[1] = __logf(se);
  }
  __syncthreads();
  if (tid < T_TAGS) out[(size_t)r * T_TAGS + tid] = sc[tid] - red[0] - red[1];
}

__global__ void init_kernel(float* h_buf, unsigned int* counter) {
  int t = blockIdx.x * blockDim.x + threadIdx.x;
  if (t < 2 * HW) h_buf[t] = 0.f;
  if (t == 0) *counter = 0u;
}

extern "C" void kernel_launch(void* const* d_in, const int* in_sizes, int n_in,
                              void* d_out, int out_size, void* d_ws, size_t ws_size,
                              hipStream_t stream) {
  const int*   sentence = (const int*)d_in[0];
  const int*   chars    = (const int*)d_in[1];
  const int*   clen     = (const int*)d_in[2];
  const float* word_emb = (const float*)d_in[3];
  const float* char_emb = (const float*)d_in[4];
  const float* Wih_c    = (const float*)d_in[5];
  const float* Whh_c    = (const float*)d_in[6];
  const float* bih_c    = (const float*)d_in[7];
  const float* bhh_c    = (const float*)d_in[8];
  const float* Wih_w    = (const float*)d_in[9];
  const float* Whh_w    = (const float*)d_in[10];
  const float* bih_w    = (const float*)d_in[11];
  const float* bhh_w    = (const float*)d_in[12];
  const float* W_tag    = (const float*)d_in[13];
  const float* b_tag    = (const float*)d_in[14];

  // workspace layout (~86 MB)
  char* ws = (char*)d_ws;
  size_t off = 0;
  float* precomp  = (float*)(ws + off); off += (size_t)S_WORDS * GW * 4;  // 64 MB
  float* h_char   = (float*)(ws + off); off += (size_t)S_WORDS * HC * 4;  //  2 MB
  float* lstm_out = (float*)(ws + off); off += (size_t)S_WORDS * HW * 4;  // 16 MB
  float* h_buf    = (float*)(ws + off); off += 2 * HW * 4;
  unsigned int* counter = (unsigned int*)(ws + off);

  init_kernel<<<4, 256, 0, stream>>>(h_buf, counter);
  char_lstm_kernel<<<S_WORDS / (CW * 16), CW * 32, CH_LDS_BYTES, stream>>>(
      chars, clen, char_emb, Wih_c, Whh_c, bih_c, bhh_c, h_char);
  wih_precompute_kernel<<<dim3(S_WORDS / 16, GW / 128), 128, 0, stream>>>(
      sentence, word_emb, h_char, Wih_w, bih_w, bhh_w, precomp);
  word_lstm_kernel<<<NWG, 512, 0, stream>>>(
      Whh_w, precomp, lstm_out, h_buf, counter);
  tag_kernel<<<S_WORDS, 64, 0, stream>>>(lstm_out, W_tag, b_tag, (float*)d_out);
}